// ParallelVarPatchEmbed_27779848471568
// MI455X (gfx1250) — compile-verified
//
#include <hip/hip_runtime.h>
#include <stdint.h>

typedef __attribute__((ext_vector_type(2))) float v2f;
typedef __attribute__((ext_vector_type(8))) float v8f;

// Padded LDS row stride (floats): 16 payload + 4 pad.
// - keeps 16B alignment for b128 LDS writes (20*P + {0,4,8,12} is mult of 4)
// - keeps v2f LDS loads 8B aligned (even element index)
// - stride-20 bank pattern is conflict-free for 16 lanes, and the +2 element
//   offset of the upper half-wave lands on a disjoint bank set (mod-4 classes).
#define LSTR 20

// CDNA5 async global->LDS copy (ASYNCcnt-tracked, no VGPR round trip).
// VDST = LDS byte address VGPR, VADDR = 64-bit global address pair, SADDR=off.
__device__ __forceinline__ void async_ld_b128(uint32_t lds_byte, const float* gaddr) {
    asm volatile("global_load_async_to_lds_b128 %0, %1, off"
                 :: "v"(lds_byte), "v"(gaddr)
                 : "memory");
}

__device__ __forceinline__ void wait_async0() {
    asm volatile("s_wait_asynccnt 0x0" ::: "memory");
}

// x           : (16, 8, 512, 512) f32
// in_vars     : (8,) int
// proj_weights: (10, 128, 1, 16, 16) f32
// proj_biases : (10, 128) f32
// out         : (16, 8, 1024, 128) f32
__global__ __launch_bounds__(256)
void patch_embed_wmma_f32(const float* __restrict__ x,
                          const int* __restrict__ in_vars,
                          const float* __restrict__ pw,
                          const float* __restrict__ pb,
                          float* __restrict__ out)
{
    __shared__ __align__(16) float sA[128 * LSTR];   // 128 patches x 16 k (padded)
    __shared__ __align__(16) float sB[128 * LSTR];   // 128 e       x 16 k (padded)

    const int tid  = threadIdx.x;
    const int lane = tid & 31;
    const int wave = tid >> 5;        // 0..7
    const int half = lane >> 4;       // 0: lanes 0-15, 1: lanes 16-31
    const int l15  = lane & 15;

    const int blk  = blockIdx.x;      // 1024 blocks
    const int pblk = blk & 7;         // patch block (128 patches each)
    const int v    = (blk >> 3) & 7;
    const int b    = blk >> 6;

    const int var = in_vars[v];
    const int ghb = pblk * 4;         // first patch-grid row of this block

    const float* xbv = x  + (size_t)(b * 8 + v) * (512 * 512);
    const float* wv  = pw + (size_t)var * (128 * 256);

    // LDS byte-offset bases (low 32 bits of the flat LDS address are the
    // wave-relative LDS offset per the CDNA5 aperture mapping).
    const uint32_t sA_base = (uint32_t)(uintptr_t)(void*)sA;
    const uint32_t sB_base = (uint32_t)(uintptr_t)(void*)sB;

    v8f acc[8] = {};                  // 16(M) x 128(N) per wave

    // K loop blocked by patch row p (K-chunk = 16)
    for (int p = 0; p < 16; ++p) {
        // ---- Stage A: 4 image rows (gh = ghb..ghb+3, row gh*16+p), 512 f32
        // each. 512 x b128 async copies, 2 per thread, fully coalesced.
        #pragma unroll
        for (int it = 0; it < 2; ++it) {
            const int g = tid + it * 256;          // vec4 index 0..511
            const int r = g >> 7;                  // which gh row (0..3)
            const int c = (g & 127) << 2;          // column 0..508
            const int P = r * 32 + (c >> 4);       // patch within block (0..127)
            const int q = c & 15;                  // k within chunk
            async_ld_b128(sA_base + 4u * (uint32_t)(P * LSTR + q),
                          xbv + (size_t)((ghb + r) * 16 + p) * 512 + c);
        }
        // ---- Stage B: weight slice W[e=0..127][p][q=0..15] -> 2048 f32.
        #pragma unroll
        for (int it = 0; it < 2; ++it) {
            const int g = tid + it * 256;          // vec4 index 0..511
            const int e = g >> 2;
            const int q = (g & 3) << 2;
            async_ld_b128(sB_base + 4u * (uint32_t)(e * LSTR + q),
                          wv + e * 256 + p * 16 + q);
        }
        wait_async0();
        __syncthreads();

        // ---- Compute: 2 kk-pairs x (1 A-pair load + 8 B-pair loads, then
        // 16 back-to-back WMMAs). Preloading all fragments lets one dscnt
        // wait cover the whole batch instead of a wait per fragment.
        const float* aRow = sA + (wave * 16 + l15) * LSTR + half * 2;
        const float* bCol = sB + l15 * LSTR + half * 2;
        #pragma unroll
        for (int kk2 = 0; kk2 < 2; ++kk2) {
            const v2f a0 = *(const v2f*)(aRow + kk2 * 8 + 0);
            const v2f a1 = *(const v2f*)(aRow + kk2 * 8 + 4);
            v2f b0[8], b1[8];
            #pragma unroll
            for (int n = 0; n < 8; ++n) {
                b0[n] = *(const v2f*)(bCol + n * 16 * LSTR + kk2 * 8 + 0);
                b1[n] = *(const v2f*)(bCol + n * 16 * LSTR + kk2 * 8 + 4);
            }
            #pragma unroll
            for (int n = 0; n < 8; ++n) {
                acc[n] = __builtin_amdgcn_wmma_f32_16x16x4_f32(
                    false, a0, false, b0[n], (short)0, acc[n], false, false);
                acc[n] = __builtin_amdgcn_wmma_f32_16x16x4_f32(
                    false, a1, false, b1[n], (short)0, acc[n], false, false);
            }
        }
        __syncthreads();
    }

    // ---- Epilogue: bias + store.
    // C/D layout: VGPR r -> M = r + 8*half, N = n*16 + l15.
    const float* bias = pb + var * 128;
    float* obase = out
        + ((size_t)(b * 8 + v) * 1024 + pblk * 128 + wave * 16 + half * 8) * 128
        + l15;
    #pragma unroll
    for (int n = 0; n < 8; ++n) {
        const float bv = bias[n * 16 + l15];
        #pragma unroll
        for (int r = 0; r < 8; ++r) {
            obase[(size_t)r * 128 + n * 16] = acc[n][r] + bv;
        }
    }
}

extern "C" void kernel_launch(void* const* d_in, const int* in_sizes, int n_in,
                              void* d_out, int out_size, void* d_ws, size_t ws_size,
                              hipStream_t stream) {
    const float* x  = (const float*)d_in[0];
    const int*   iv = (const int*)d_in[1];
    const float* pw = (const float*)d_in[2];
    const float* pb = (const float*)d_in[3];
    float* out = (float*)d_out;

    // 16 (b) * 8 (v) * 8 (patch blocks of 128) = 1024 workgroups, 8 waves each.
    patch_embed_wmma_f32<<<dim3(1024), dim3(256), 0, stream>>>(x, iv, pw, pb, out);
}